// STBlock_38783554683504
// MI455X (gfx1250) — compile-verified
//
#include <hip/hip_runtime.h>

typedef __attribute__((ext_vector_type(2))) float v2f;
typedef __attribute__((ext_vector_type(8))) float v8f;

// ---------------------------------------------------------------- utilities

__global__ void zero_f(float* __restrict__ p, int n) {
  int i = blockIdx.x * blockDim.x + threadIdx.x;
  if (i < n) p[i] = 0.f;
}

__global__ void zero_f4(float4* __restrict__ p, int n4) {
  int i = blockIdx.x * blockDim.x + threadIdx.x;
  if (i < n4) p[i] = make_float4(0.f, 0.f, 0.f, 0.f);
}

__global__ void deg_kernel(const int* __restrict__ src, const int* __restrict__ dst,
                           float* __restrict__ out_deg, float* __restrict__ in_deg, int E) {
  int e = blockIdx.x * blockDim.x + threadIdx.x;
  if (e < E) {
    atomicAdd(&out_deg[src[e]], 1.f);
    atomicAdd(&in_deg[dst[e]], 1.f);
  }
}

__global__ void norm_kernel(float* __restrict__ p, int n) {
  int i = blockIdx.x * blockDim.x + threadIdx.x;
  if (i < n) {
    float d = p[i];
    d = d < 1.f ? 1.f : d;
    p[i] = rsqrtf(d);   // clip(deg,1)^-0.5
  }
}

// h0_pre[n][t][f] = tf[n][f][t] * out_norm[n]   (F_IN=32, T=8)
__global__ void trans_scale(const float* __restrict__ tf, const float* __restrict__ out_norm,
                            float* __restrict__ h0, int total /* N*256 */) {
  int gid = blockIdx.x * blockDim.x + threadIdx.x;
  if (gid >= total) return;
  int n = gid >> 8;
  int r = gid & 255;
  int t = r >> 5;
  int f = r & 31;
  h0[gid] = tf[(n << 8) + (f << 3) + t] * out_norm[n];
}

// agg[dst[e]][:] += hpre[src[e]][:]   (RFQ = row_floats/4; one thread = one float4)
template<int RFQ>
__global__ void scatter_add(const float* __restrict__ hpre, const int* __restrict__ src,
                            const int* __restrict__ dst, float* __restrict__ agg, int total) {
  int gid = blockIdx.x * blockDim.x + threadIdx.x;
  if (gid >= total) return;
  int e = gid / RFQ;
  int q = gid - e * RFQ;
  int s = src[e], d = dst[e];
  const float4 v = ((const float4*)(hpre + (size_t)s * (RFQ * 4)))[q];
  float* p = agg + (size_t)d * (RFQ * 4) + (size_t)q * 4;
  atomicAdd(p + 0, v.x);
  atomicAdd(p + 1, v.y);
  atomicAdd(p + 2, v.z);
  atomicAdd(p + 3, v.w);
}

// ---------------------------------------------------------------- WMMA GEMM
// Y[row, col] = act( (in_norm[row/8] * X[row,:]) @ W[:,col] + bias[col] ) * (out_norm[row/8])?
// One wave per 16x16 tile; V_WMMA_F32_16X16X4_F32, K stepped by 4.
// A layout (ISA 16x4 f32): lane l -> M = l&15, K pair {2*(l>>4), 2*(l>>4)+1}.
// B symmetric; staged in LDS as [n][K] so each step is an aligned 8B ds read.

template<int K_DIM, int N_DIM, bool RELU, bool SCALE_OUT>
__global__ __launch_bounds__(256)
void gemm_wmma(const float* __restrict__ X, const float* __restrict__ W,
               const float* __restrict__ bias, const float* __restrict__ in_norm,
               const float* __restrict__ out_norm, float* __restrict__ Y, int m_tiles) {
  __shared__ float wlds[16 * K_DIM];
  __shared__ float blds[16];
  const int obase = blockIdx.y * 16;

  for (int idx = threadIdx.x; idx < 16 * K_DIM; idx += blockDim.x) {
    int nl = idx / K_DIM;
    int k  = idx - nl * K_DIM;
    int col = obase + nl;
    wlds[idx] = (col < N_DIM) ? W[(size_t)k * N_DIM + col] : 0.f;
  }
  if (threadIdx.x < 16) {
    int col = obase + (int)threadIdx.x;
    blds[threadIdx.x] = (col < N_DIM) ? bias[col] : 0.f;
  }
  __syncthreads();

  const int wave = threadIdx.x >> 5;
  const int lane = threadIdx.x & 31;
  const int tile = blockIdx.x * 8 + wave;
  if (tile >= m_tiles) return;

  const int m    = lane & 15;
  const int half = lane >> 4;
  const int rowA = tile * 16 + m;
  const float scale = in_norm[rowA >> 3];              // T == 8
  const float* xrow = X + (size_t)rowA * K_DIM;

  v8f c = {};
#pragma unroll
  for (int k0 = 0; k0 < K_DIM; k0 += 4) {
    const int ka = k0 + 2 * half;
    v2f a = *(const v2f*)(xrow + ka);
    a = a * scale;                                     // fold in_norm into A
    v2f b = *(const v2f*)(&wlds[m * K_DIM + ka]);
    c = __builtin_amdgcn_wmma_f32_16x16x4_f32(false, a, false, b, (short)0, c, false, false);
  }

#pragma unroll
  for (int r = 0; r < 8; ++r) {
    const int rowo = tile * 16 + r + 8 * half;         // D: vgpr r -> M = r (+8 for hi half)
    const int col  = obase + m;
    float v = c[r] + blds[m];
    if (RELU) v = fmaxf(v, 0.f);
    if (SCALE_OUT) v = v * out_norm[rowo >> 3];        // fold next layer's out_norm
    if (col < N_DIM) Y[(size_t)rowo * N_DIM + col] = v;
  }
}

// ---------------------------------------------------------------- Conv1d(k=3,pad=1) as K=192 GEMM
// out[n,o,t] = sum_{tap,i} h3[n, t+tap-1, i] * Wc[o,i,tap] + bc[o]
__global__ __launch_bounds__(256)
void conv_wmma(const float* __restrict__ h3, const float* __restrict__ Wc,
               const float* __restrict__ bc, float* __restrict__ out, int m_tiles) {
  __shared__ float wlds[16 * 192];
  __shared__ float blds[16];
  const int obase = blockIdx.y * 16;

  for (int idx = threadIdx.x; idx < 16 * 192; idx += blockDim.x) {
    int ol = idx / 192;
    int k  = idx - ol * 192;
    int tap = k >> 6;
    int i   = k & 63;
    wlds[idx] = Wc[((size_t)(obase + ol) * 64 + i) * 3 + tap];
  }
  if (threadIdx.x < 16) blds[threadIdx.x] = bc[obase + threadIdx.x];
  __syncthreads();

  const int wave = threadIdx.x >> 5;
  const int lane = threadIdx.x & 31;
  const int tile = blockIdx.x * 8 + wave;
  if (tile >= m_tiles) return;

  const int m    = lane & 15;
  const int half = lane >> 4;
  const int rowA = tile * 16 + m;
  const int nA = rowA >> 3;
  const int tA = rowA & 7;

  v8f c = {};
#pragma unroll
  for (int k0 = 0; k0 < 192; k0 += 4) {
    const int ka  = k0 + 2 * half;
    const int tap = ka >> 6;           // ka, ka+1 always in same 64-block (ka even)
    const int i   = ka & 63;
    const int ts  = tA + tap - 1;
    v2f a; a.x = 0.f; a.y = 0.f;
    if ((unsigned)ts < 8u)             // zero padding at t boundaries
      a = *(const v2f*)(h3 + ((size_t)(nA * 8 + ts) * 64 + i));
    v2f b = *(const v2f*)(&wlds[m * 192 + ka]);
    c = __builtin_amdgcn_wmma_f32_16x16x4_f32(false, a, false, b, (short)0, c, false, false);
  }

#pragma unroll
  for (int r = 0; r < 8; ++r) {
    const int rowo = tile * 16 + r + 8 * half;
    const int o = obase + m;
    const int n = rowo >> 3;
    const int t = rowo & 7;
    out[((size_t)n * 64 + o) * 8 + t] = c[r] + blds[m];   // [N, F_OUT, T] layout
  }
}

// ---------------------------------------------------------------- launcher

extern "C" void kernel_launch(void* const* d_in, const int* in_sizes, int n_in,
                              void* d_out, int out_size, void* d_ws, size_t ws_size,
                              hipStream_t stream) {
  const float* tf  = (const float*)d_in[0];
  const int*   src = (const int*)d_in[1];
  const int*   dst = (const int*)d_in[2];
  const float* W1  = (const float*)d_in[3];
  const float* b1  = (const float*)d_in[4];
  const float* W2  = (const float*)d_in[5];
  const float* b2  = (const float*)d_in[6];
  const float* W3  = (const float*)d_in[7];
  const float* b3  = (const float*)d_in[8];
  const float* Wc  = (const float*)d_in[9];
  const float* bc  = (const float*)d_in[10];
  float* out = (float*)d_out;
  (void)n_in; (void)out_size; (void)ws_size;

  const int F_IN = 32, T = 8, H1 = 40, H2 = 64;
  const int N = in_sizes[0] / (F_IN * T);     // 50000
  const int E = in_sizes[1];                  // 800000
  const int M = N * T;                        // 400000 rows
  const int m_tiles = M / 16;                 // 25000
  const int gx = (m_tiles + 7) / 8;

  float* ws = (float*)d_ws;
  float* out_norm = ws;                       // reused: out_deg -> out_norm
  float* in_norm  = ws + N;                   // reused: in_deg  -> in_norm
  size_t base = ((size_t)2 * N + 3) & ~(size_t)3;
  float* slotA = ws + base;                   // M*64 floats each slot
  float* slotB = slotA + (size_t)M * H2;

  const int THR = 256;
  auto blocks = [](int n) { return (n + 255) / 256; };

  // degrees -> norms
  zero_f<<<blocks(2 * N), THR, 0, stream>>>(out_norm, 2 * N);
  deg_kernel<<<blocks(E), THR, 0, stream>>>(src, dst, out_norm, in_norm, E);
  norm_kernel<<<blocks(2 * N), THR, 0, stream>>>(out_norm, 2 * N);

  // layer-0 input: transpose [N,F,T]->[N,T,F] with out_norm folded
  trans_scale<<<blocks(N * 256), THR, 0, stream>>>(tf, out_norm, slotA, N * 256);

  // layer 1: scatter (256 f/row) then GEMM K=32 -> 40, relu, *out_norm
  zero_f4<<<blocks(M * F_IN / 4), THR, 0, stream>>>((float4*)slotB, M * F_IN / 4);
  scatter_add<64><<<blocks(E * 64), THR, 0, stream>>>(slotA, src, dst, slotB, E * 64);
  gemm_wmma<32, 40, true, true><<<dim3(gx, 3), THR, 0, stream>>>(
      slotB, W1, b1, in_norm, out_norm, slotA, m_tiles);

  // layer 2: scatter (320 f/row) then GEMM K=40 -> 64, relu, *out_norm
  zero_f4<<<blocks(M * H1 / 4), THR, 0, stream>>>((float4*)slotB, M * H1 / 4);
  scatter_add<80><<<blocks(E * 80), THR, 0, stream>>>(slotA, src, dst, slotB, E * 80);
  gemm_wmma<40, 64, true, true><<<dim3(gx, 4), THR, 0, stream>>>(
      slotB, W2, b2, in_norm, out_norm, slotA, m_tiles);

  // layer 3: scatter (512 f/row) then GEMM K=64 -> 64, no act, no out-scale
  zero_f4<<<blocks(M * H2 / 4), THR, 0, stream>>>((float4*)slotB, M * H2 / 4);
  scatter_add<128><<<blocks(E * 128), THR, 0, stream>>>(slotA, src, dst, slotB, E * 128);
  gemm_wmma<64, 64, false, false><<<dim3(gx, 4), THR, 0, stream>>>(
      slotB, W3, b3, in_norm, out_norm, slotA, m_tiles);

  // conv1d over T (K=192 GEMM) + bias -> d_out [N, 64, 8]
  conv_wmma<<<dim3(gx, 4), THR, 0, stream>>>(slotA, Wc, bc, out, m_tiles);
}